// QweNet_21500606284503
// MI455X (gfx1250) — compile-verified
//
#include <hip/hip_runtime.h>

#define D_FEAT 128
#define D_HID  64
#define NEG_SLOPE 0.01f

typedef __attribute__((ext_vector_type(2))) float v2f;
typedef __attribute__((ext_vector_type(8))) float v8f;

__device__ __forceinline__ float leaky(float v) {
    return v > 0.0f ? v : NEG_SLOPE * v;
}

// ---------------------------------------------------------------------------
// Degree: deg[col[e]] += 1.0  (deg buffer pre-zeroed)
// ---------------------------------------------------------------------------
__global__ __launch_bounds__(256)
void degree_kernel(const long long* __restrict__ col, float* __restrict__ deg,
                   long long E) {
    long long e = (long long)blockIdx.x * blockDim.x + threadIdx.x;
    if (e < E) atomicAdd(&deg[(int)col[e]], 1.0f);
}

// dinv[i] = rsqrt(deg[i] + 1.0)   (in place)
__global__ __launch_bounds__(256)
void dinv_kernel(float* __restrict__ d, int N) {
    int i = blockIdx.x * blockDim.x + threadIdx.x;
    if (i < N) d[i] = rsqrtf(d[i] + 1.0f);
}

// ---------------------------------------------------------------------------
// WMMA f32 GEMM:  Y[N x 128] = X[N x 128] @ W[128 x 128]
// grid.x = ceil(N/16) row tiles, block = 256 threads = 8 waves.
// Wave w computes the 16x16 output tile at columns [16w, 16w+16).
// A-frag (16x4, ISA 7.12.2): lanes 0-15 hold M=lane, VGPR{0,1}=K{k0,k0+1};
//                            lanes 16-31 hold M=lane-16, VGPR{0,1}=K{k0+2,k0+3}.
// B-frag (4x16): N striped across lanes, K across VGPRs/half-wave (mirror of A).
// C/D (16x16): VGPR r -> M=r (lanes 0-15) / M=8+r (lanes 16-31), N=lane%16.
// ---------------------------------------------------------------------------
__global__ __launch_bounds__(256)
void gemm128_wmma(const float* __restrict__ X, const float* __restrict__ W,
                  float* __restrict__ Y, int N) {
    const int lane = threadIdx.x & 31;
    const int wave = threadIdx.x >> 5;
    const int lid  = lane & 15;
    const int half = lane >> 4;            // 0: lanes 0-15, 1: lanes 16-31

    int row0 = blockIdx.x * 16;
    if (row0 + 16 > N) row0 = N - 16;      // clamp: duplicate identical writes
    const int col0 = wave * 16;

    const float* xrow = X + (size_t)(row0 + lid) * D_FEAT + half * 2;
    const float* wcol = W + (size_t)(half * 2) * D_FEAT + col0 + lid;

    v8f acc = {};
#pragma unroll
    for (int k0 = 0; k0 < D_FEAT; k0 += 4) {
        v2f a, b;
        a.x = xrow[k0 + 0];                         // A[M=lid][k0 + 2*half + 0]
        a.y = xrow[k0 + 1];                         // A[M=lid][k0 + 2*half + 1]
        b.x = wcol[(size_t)(k0 + 0) * D_FEAT];      // B[k0 + 2*half + 0][col0+lid]
        b.y = wcol[(size_t)(k0 + 1) * D_FEAT];      // B[k0 + 2*half + 1][col0+lid]
        acc = __builtin_amdgcn_wmma_f32_16x16x4_f32(
                  false, a, false, b, (short)0, acc, false, false);
    }

    float* yout = Y + (size_t)(row0 + half * 8) * D_FEAT + col0 + lid;
#pragma unroll
    for (int r = 0; r < 8; ++r)
        yout[(size_t)r * D_FEAT] = acc[r];
}

// ---------------------------------------------------------------------------
// Edge scatter-add: agg[col] += H[row] * (dinv[row]*dinv[col])
// One wave (32 lanes) per edge; float4 per lane covers the 128-dim row.
// agg pre-zeroed. All traffic L2-resident (H/agg ~51MB each, L2 = 192MB).
// ---------------------------------------------------------------------------
__global__ __launch_bounds__(256)
void edge_aggregate(const float* __restrict__ H,
                    const long long* __restrict__ row,
                    const long long* __restrict__ col,
                    const float* __restrict__ dinv,
                    float* __restrict__ agg, long long E) {
    const int lane = threadIdx.x & 31;
    long long e = (long long)blockIdx.x * 8 + (threadIdx.x >> 5);
    if (e >= E) return;
    const int r = (int)row[e];
    const int c = (int)col[e];
    const float nrm = dinv[r] * dinv[c];
    const float4 v = ((const float4*)(H + (size_t)r * D_FEAT))[lane];
    float* ap = agg + (size_t)c * D_FEAT + lane * 4;
    atomicAdd(ap + 0, v.x * nrm);
    atomicAdd(ap + 1, v.y * nrm);
    atomicAdd(ap + 2, v.z * nrm);
    atomicAdd(ap + 3, v.w * nrm);
}

// ---------------------------------------------------------------------------
// out = leaky_relu(agg + h * dinv^2 + bias)   (one thread per element)
// ---------------------------------------------------------------------------
__global__ __launch_bounds__(256)
void combine_kernel(const float* __restrict__ agg, const float* __restrict__ h,
                    const float* __restrict__ dinv, const float* __restrict__ bias,
                    float* __restrict__ out, int N) {
    size_t idx = (size_t)blockIdx.x * blockDim.x + threadIdx.x;
    if (idx >= (size_t)N * D_FEAT) return;
    const int node = (int)(idx >> 7);
    const int d    = (int)(idx & (D_FEAT - 1));
    const float di = dinv[node];
    out[idx] = leaky(agg[idx] + h[idx] * di * di + bias[d]);
}

// ---------------------------------------------------------------------------
// MLP head: out[n] = leaky(H[n] @ Wd1 + bd1) @ Wd2 + bd2
// One wave per node; row staged in LDS; each lane owns 2 of the 64 hidden
// units; __shfl_down wave reduction for the final length-64 dot.
// ---------------------------------------------------------------------------
__global__ __launch_bounds__(256)
void head_kernel(const float* __restrict__ H, const float* __restrict__ Wd1,
                 const float* __restrict__ bd1, const float* __restrict__ Wd2,
                 const float* __restrict__ bd2, float* __restrict__ out, int N) {
    __shared__ float sh[8][D_FEAT];
    const int lane = threadIdx.x & 31;
    const int wave = threadIdx.x >> 5;
    int node = blockIdx.x * 8 + wave;
    if (node >= N) node = N - 1;           // clamp: duplicate identical writes

    ((float4*)sh[wave])[lane] = ((const float4*)(H + (size_t)node * D_FEAT))[lane];
    __syncthreads();

    const int j0 = lane;
    const int j1 = lane + 32;
    float t0 = bd1[j0];
    float t1 = bd1[j1];
#pragma unroll 8
    for (int k = 0; k < D_FEAT; ++k) {
        const float hv = sh[wave][k];
        t0 += hv * Wd1[k * D_HID + j0];
        t1 += hv * Wd1[k * D_HID + j1];
    }
    float p = leaky(t0) * Wd2[j0] + leaky(t1) * Wd2[j1];
#pragma unroll
    for (int off = 16; off > 0; off >>= 1)
        p += __shfl_down(p, off, 32);
    if (lane == 0) out[node] = p + bd2[0];
}

// ---------------------------------------------------------------------------
extern "C" void kernel_launch(void* const* d_in, const int* in_sizes, int n_in,
                              void* d_out, int out_size, void* d_ws, size_t ws_size,
                              hipStream_t stream) {
    (void)n_in; (void)out_size; (void)ws_size;

    const float*     x   = (const float*)d_in[0];
    const long long* ei  = (const long long*)d_in[1];   // int64 edge_index [2, E]
    const float*     W1  = (const float*)d_in[2];
    const float*     b1  = (const float*)d_in[3];
    const float*     W2  = (const float*)d_in[4];
    const float*     b2  = (const float*)d_in[5];
    const float*     Wd1 = (const float*)d_in[6];
    const float*     Wd2 = (const float*)d_in[8];
    const float*     bd1 = (const float*)d_in[7];
    const float*     bd2 = (const float*)d_in[9];
    float*           out = (float*)d_out;

    const int       N = in_sizes[0] / D_FEAT;
    const long long E = (long long)in_sizes[1] / 2;
    const long long* row = ei;
    const long long* col = ei + E;

    // Workspace layout (~154 MB): bufA | bufB | agg | dinv
    float* bufA = (float*)d_ws;
    float* bufB = bufA + (size_t)N * D_FEAT;
    float* agg  = bufB + (size_t)N * D_FEAT;
    float* dinv = agg  + (size_t)N * D_FEAT;

    const int rowTiles  = (N + 15) / 16;
    const int edgeBlks  = (int)((E + 7) / 8);
    const int elemBlks  = (int)(((size_t)N * D_FEAT + 255) / 256);
    const int nodeBlks  = (N + 7) / 8;

    // --- degree / dinv (shared by both conv layers) ---
    hipMemsetAsync(dinv, 0, (size_t)N * sizeof(float), stream);
    degree_kernel<<<(int)((E + 255) / 256), 256, 0, stream>>>(col, dinv, E);
    dinv_kernel<<<(N + 255) / 256, 256, 0, stream>>>(dinv, N);

    // --- GCN layer 1 ---
    gemm128_wmma<<<rowTiles, 256, 0, stream>>>(x, W1, bufA, N);
    hipMemsetAsync(agg, 0, (size_t)N * D_FEAT * sizeof(float), stream);
    edge_aggregate<<<edgeBlks, 256, 0, stream>>>(bufA, row, col, dinv, agg, E);
    combine_kernel<<<elemBlks, 256, 0, stream>>>(agg, bufA, dinv, b1, bufB, N);

    // --- GCN layer 2 ---
    gemm128_wmma<<<rowTiles, 256, 0, stream>>>(bufB, W2, bufA, N);
    hipMemsetAsync(agg, 0, (size_t)N * D_FEAT * sizeof(float), stream);
    edge_aggregate<<<edgeBlks, 256, 0, stream>>>(bufA, row, col, dinv, agg, E);
    combine_kernel<<<elemBlks, 256, 0, stream>>>(agg, bufA, dinv, b2, bufB, N);

    // --- MLP head ---
    head_kernel<<<nodeBlks, 256, 0, stream>>>(bufB, Wd1, bd1, Wd2, bd2, out, N);
}